// NLLvMF_60120952209635
// MI455X (gfx1250) — compile-verified
//
#include <hip/hip_runtime.h>
#include <hip/hip_bf16.h>

#define NROWS 8192   // BATCH*SEQ = 32*256
#define DIM   512    // EMB_DIM
#define PAD_ID 1
#define N_TERMS 256

typedef __attribute__((ext_vector_type(2))) float v2f;
typedef __attribute__((ext_vector_type(8))) float v8f;

// ---------------------------------------------------------------------------
// Kernel 1: per-row dot(preds, emb[target]) and ||preds||^2 via fp32 WMMA.
// One wave handles 16 rows. A-fragment (16x4 f32, 2 VGPRs): lane L<16 holds
// rows' K=0,1; lane L+16 holds K=2,3. B-fragment (4x16 f32) has the symmetric
// layout, so the same addressing builds both fragments, and the preds
// fragment can be reused as both A and B for the norm WMMA.
// Diagonal of the 16x16 f32 accumulator = the 16 per-row reductions.
// ---------------------------------------------------------------------------
__global__ void __launch_bounds__(256)
nllvmf_dot_kernel(const float* __restrict__ preds,
                  const int*   __restrict__ target,
                  const float* __restrict__ emb,
                  float* __restrict__ dotv,
                  float* __restrict__ z2v) {
    const int tid  = blockIdx.x * blockDim.x + threadIdx.x;
    const int wave = tid >> 5;              // 512 waves, 16 rows each
    const int lane = threadIdx.x & 31;
    const int r0   = wave << 4;
    const int rsel = lane & 15;
    const int half = lane >> 4;             // 0: K={0,1}, 1: K={2,3}
    const int row  = r0 + rsel;

    const int tg = target[row];
    const float* __restrict__ pA = preds + (size_t)row * DIM + 2 * half;
    const float* __restrict__ pB = emb   + (size_t)tg  * DIM + 2 * half;

    v8f cdot = {};
    v8f cnrm = {};
#pragma unroll 8
    for (int kk = 0; kk < DIM; kk += 4) {
        v2f a = *(const v2f*)(pA + kk);   // preds fragment (A and B layout)
        v2f b = *(const v2f*)(pB + kk);   // gathered embedding fragment (B)
        cdot = __builtin_amdgcn_wmma_f32_16x16x4_f32(
                   false, a, false, b, (short)0, cdot, false, false);
        cnrm = __builtin_amdgcn_wmma_f32_16x16x4_f32(
                   false, a, false, a, (short)0, cnrm, false, false);
    }

    // Diagonal D[i][i]: i<8 -> VGPR i, lane i ; i>=8 -> VGPR i-8, lane i+16.
    int idx = 0, orow = 0;
    bool valid = false;
    if (lane < 8)        { idx = lane;      orow = r0 + lane;      valid = true; }
    else if (lane >= 24) { idx = lane - 24; orow = r0 + lane - 16; valid = true; }

    float dv = 0.0f, nv = 0.0f;
#pragma unroll
    for (int q = 0; q < 8; ++q) {
        if (idx == q) { dv = cdot[q]; nv = cnrm[q]; }
    }
    if (valid) {
        dotv[orow] = dv;
        z2v[orow]  = nv;
    }
}

// ---------------------------------------------------------------------------
// Kernel 2: per-row loss. One wave per row; lane j evaluates series terms
// j, j+32, ..., j+224; wave-wide logsumexp via lane shuffles.
//   loss = logS - z - dot + [256 log(2*pi) - 255 log 2 - lgamma(256)]
// ---------------------------------------------------------------------------
__global__ void __launch_bounds__(256)
nllvmf_loss_kernel(const float* __restrict__ dotv,
                   const float* __restrict__ z2v,
                   const int*   __restrict__ target,
                   float* __restrict__ lossArr) {
    const int gtid = blockIdx.x * blockDim.x + threadIdx.x;
    const int row  = gtid >> 5;             // one wave (32 lanes) per row
    const int lane = threadIdx.x & 31;
    if (row >= NROWS) return;

    const float z2   = z2v[row];
    const float z    = sqrtf(z2);
    const float logx = logf(0.25f * z2);    // log(z^2/4)
    const float lg256 = lgammaf(256.0f);

    float t[N_TERMS / 32];
    float m = -__builtin_inff();
#pragma unroll
    for (int q = 0; q < N_TERMS / 32; ++q) {
        const float j = (float)(lane + 32 * q);
        const float term = j * logx - lgammaf(j + 1.0f)
                         - (lgammaf(256.0f + j) - lg256);
        t[q] = term;
        m = fmaxf(m, term);
    }
    for (int off = 16; off > 0; off >>= 1)
        m = fmaxf(m, __shfl_xor(m, off, 32));

    float s = 0.0f;
#pragma unroll
    for (int q = 0; q < N_TERMS / 32; ++q)
        s += expf(t[q] - m);
    for (int off = 16; off > 0; off >>= 1)
        s += __shfl_xor(s, off, 32);

    if (lane == 0) {
        const float logS = m + logf(s);
        const float CONSTT = 256.0f * logf(6.283185307179586f)
                           - 255.0f * logf(2.0f) - lg256;
        const float loss = logS - z - dotv[row] + CONSTT;
        lossArr[row] = (target[row] != PAD_ID) ? loss : 0.0f;
    }
}

// ---------------------------------------------------------------------------
// Kernel 3: deterministic final reduction (fixed strided order + LDS tree).
// out = sum(loss * mask) / sum(mask)
// ---------------------------------------------------------------------------
__global__ void __launch_bounds__(256)
nllvmf_reduce_kernel(const float* __restrict__ lossArr,
                     const int*   __restrict__ target,
                     float* __restrict__ out) {
    __shared__ float sl[256];
    __shared__ float sc[256];
    const int tid = threadIdx.x;
    float a = 0.0f, c = 0.0f;
    for (int i = tid; i < NROWS; i += 256) {
        a += lossArr[i];
        c += (target[i] != PAD_ID) ? 1.0f : 0.0f;
    }
    sl[tid] = a;
    sc[tid] = c;
    __syncthreads();
    for (int off = 128; off > 0; off >>= 1) {
        if (tid < off) {
            sl[tid] += sl[tid + off];
            sc[tid] += sc[tid + off];
        }
        __syncthreads();
    }
    if (tid == 0) out[0] = sl[0] / sc[0];
}

extern "C" void kernel_launch(void* const* d_in, const int* in_sizes, int n_in,
                              void* d_out, int out_size, void* d_ws, size_t ws_size,
                              hipStream_t stream) {
    const float* preds  = (const float*)d_in[0];   // [32,256,512] f32
    const int*   target = (const int*)d_in[1];     // [32,256] int
    const float* emb    = (const float*)d_in[2];   // [50257,512] f32

    float* dotv    = (float*)d_ws;                 // [8192]
    float* z2v     = dotv + NROWS;                 // [8192]
    float* lossArr = z2v + NROWS;                  // [8192]

    // 512 waves x 16 rows = 8192 rows; 8 waves per 256-thread block.
    nllvmf_dot_kernel<<<NROWS / 128, 256, 0, stream>>>(preds, target, emb, dotv, z2v);
    // One wave per row: 8192 waves = 1024 blocks of 256 threads.
    nllvmf_loss_kernel<<<NROWS / 8, 256, 0, stream>>>(dotv, z2v, target, lossArr);
    nllvmf_reduce_kernel<<<1, 256, 0, stream>>>(lossArr, target, (float*)d_out);
}